// SemanticBusArchitecture_10393820856411
// MI455X (gfx1250) — compile-verified
//
#include <hip/hip_runtime.h>

// Problem constants (match reference)
#define N_MODULES   32
#define POINTER_DIM 64
#define CONTENT_DIM 256
#define BATCH       16
#define TOKENS      128
#define RIN_DIM     (CONTENT_DIM + POINTER_DIM)   // 320

typedef __attribute__((ext_vector_type(2))) float v2f;
typedef __attribute__((ext_vector_type(4))) float v4f;
typedef __attribute__((ext_vector_type(8))) float v8f;

// d_out layout (floats): out_p [B,M,P], out_c [B,M,C], mems [B,M,C,C]
#define OUT_P_OFF 0
#define OUT_C_OFF (BATCH * N_MODULES * POINTER_DIM)                    // 32768
#define MEMS_OFF  (OUT_C_OFF + BATCH * N_MODULES * CONTENT_DIM)       // 163840

// d_ws layout (floats)
#define WS_EFFC  0
#define WS_EFFP  (WS_EFFC  + BATCH * N_MODULES * CONTENT_DIM)         // 131072
#define WS_EOUTC (WS_EFFP  + BATCH * N_MODULES * POINTER_DIM)         // 163840
#define WS_INV   (WS_EOUTC + BATCH * N_MODULES * CONTENT_DIM)         // 294912

// ---------------------------------------------------------------------------
// K1: per batch b -> w_in (LDS), eff_c, eff_p (workspace)
// ---------------------------------------------------------------------------
__global__ __launch_bounds__(256) void k1_gather(
    const float* __restrict__ pointers,   // [B,T,P]
    const float* __restrict__ contents,   // [B,T,C]
    const float* __restrict__ receivers,  // [M,P]
    float* __restrict__ ws) {
  const int b   = blockIdx.x;
  const int tid = threadIdx.x;
  __shared__ float w_in[N_MODULES * TOKENS];  // 16 KB

  // w_in[m][t] = relu(tanh(receivers[m] . pointers[b,t]))
  for (int j = 0; j < (N_MODULES * TOKENS) / 256; ++j) {
    int idx = tid + j * 256;
    int m = idx >> 7;
    int t = idx & 127;
    const float* rp = receivers + m * POINTER_DIM;
    const float* pp = pointers + ((size_t)b * TOKENS + t) * POINTER_DIM;
    float acc = 0.f;
    #pragma unroll 8
    for (int p = 0; p < POINTER_DIM; ++p) acc += rp[p] * pp[p];
    float w = tanhf(acc);
    w_in[idx] = w > 0.f ? w : 0.f;
  }
  __syncthreads();

  // eff_c[m][c] = sum_t w_in[m][t] * contents[b,t,c]   (thread owns column c)
  {
    const int c = tid;
    float acc[N_MODULES];
    #pragma unroll
    for (int m = 0; m < N_MODULES; ++m) acc[m] = 0.f;
    const float* cb = contents + (size_t)b * TOKENS * CONTENT_DIM + c;
    for (int t = 0; t < TOKENS; ++t) {
      float v = cb[(size_t)t * CONTENT_DIM];
      #pragma unroll
      for (int m = 0; m < N_MODULES; ++m) acc[m] += w_in[m * TOKENS + t] * v;
    }
    float* ec = ws + WS_EFFC + (size_t)b * N_MODULES * CONTENT_DIM + c;
    #pragma unroll
    for (int m = 0; m < N_MODULES; ++m) ec[m * CONTENT_DIM] = acc[m];
  }

  // eff_p[m][p] = sum_t w_in[m][t] * pointers[b,t,p]
  {
    const int p = tid & 63;
    const int g = tid >> 6;   // quarter of modules
    float acc[8];
    #pragma unroll
    for (int j = 0; j < 8; ++j) acc[j] = 0.f;
    const float* pb = pointers + (size_t)b * TOKENS * POINTER_DIM + p;
    for (int t = 0; t < TOKENS; ++t) {
      float v = pb[(size_t)t * POINTER_DIM];
      #pragma unroll
      for (int j = 0; j < 8; ++j) acc[j] += w_in[(g * 8 + j) * TOKENS + t] * v;
    }
    float* ep = ws + WS_EFFP + (size_t)b * N_MODULES * POINTER_DIM + p;
    #pragma unroll
    for (int j = 0; j < 8; ++j) ep[(g * 8 + j) * POINTER_DIM] = acc[j];
  }
}

// ---------------------------------------------------------------------------
// K2: per module m -> out_c[:,m,:] = eff_c[:,m,:] (16x256) x memories[m]^T (256x256)
// via V_WMMA_F32_16X16X4_F32. 16 waves/block, one 16x16 output tile per wave.
// ---------------------------------------------------------------------------
__global__ __launch_bounds__(512) void k2_memread(
    const float* __restrict__ memories,   // [M,C,C]
    const float* __restrict__ ws,
    float* __restrict__ out) {
  const int m    = blockIdx.x;
  const int tid  = threadIdx.x;
  const int lane = tid & 31;
  const int wave = tid >> 5;               // 0..15 -> N tile
  __shared__ float sA[BATCH * CONTENT_DIM];  // eff_c[:,m,:] 16 KB

  const float* effc = ws + WS_EFFC;
  for (int j = 0; j < (BATCH * CONTENT_DIM) / 512; ++j) {
    int e = tid + j * 512;
    int row = e >> 8;       // batch row
    int d   = e & 255;      // k index
    sA[e] = effc[((size_t)row * N_MODULES + m) * CONTENT_DIM + d];
  }
  __syncthreads();

  const int n0   = wave * 16;
  const int hi   = lane >> 4;         // 0: lanes 0-15, 1: lanes 16-31
  const int col  = lane & 15;         // A row (=batch) and B column
  const int koff = hi * 2;            // K sub-offset per A/B f32 layout
  // B[k][n] = memories[m, n0+n, k]  (memories^T)
  const float* Brow = memories + ((size_t)m * CONTENT_DIM + (n0 + col)) * CONTENT_DIM + koff;
  const float* Arow = sA + (size_t)col * CONTENT_DIM + koff;

  v8f acc = {};
  for (int k = 0; k < CONTENT_DIM; k += 4) {
    v2f a;
    a.x = Arow[k];
    a.y = Arow[k + 1];
    v2f bv = *(const v2f*)(Brow + k);
    // D = A(16x4) * B(4x16) + C, full f32
    acc = __builtin_amdgcn_wmma_f32_16x16x4_f32(
        /*neg_a=*/false, a, /*neg_b=*/false, bv,
        /*c_mod=*/(short)0, acc, /*reuse_a=*/false, /*reuse_b=*/false);
  }

  // D layout: VGPR r -> (M=r, N=lane) lanes 0-15 ; (M=8+r, N=lane-16) lanes 16-31
  float* oc = out + OUT_C_OFF;
  const int brow_base = hi * 8;
  #pragma unroll
  for (int r = 0; r < 8; ++r) {
    int brow = brow_base + r;   // batch index
    oc[((size_t)brow * N_MODULES + m) * CONTENT_DIM + n0 + col] = acc[r];
  }
}

// ---------------------------------------------------------------------------
// K3: per batch b -> routing MLP, out_p, w_out, eff_out_c, inv norms
// ---------------------------------------------------------------------------
__global__ __launch_bounds__(256) void k3_route(
    const float* __restrict__ W1,  // [M,320,64]
    const float* __restrict__ b1,  // [M,64]
    const float* __restrict__ W2,  // [M,64,64]
    const float* __restrict__ b2,  // [M,64]
    float* __restrict__ ws,
    float* __restrict__ out) {
  const int b   = blockIdx.x;
  const int tid = threadIdx.x;
  __shared__ float sP[N_MODULES * POINTER_DIM];    // eff_p        8 KB
  __shared__ float sBig[N_MODULES * CONTENT_DIM];  // eff_c/out_c 32 KB
  __shared__ float sH[N_MODULES * POINTER_DIM];    // h            8 KB
  __shared__ float sOP[N_MODULES * POINTER_DIM];   // out_p        8 KB
  __shared__ float sW[N_MODULES * N_MODULES];      // w_out        4 KB

  for (int j = 0; j < 8; ++j) {
    int e = tid + j * 256;
    sP[e] = ws[WS_EFFP + (size_t)b * N_MODULES * POINTER_DIM + e];
  }
  for (int j = 0; j < 32; ++j) {
    int e = tid + j * 256;
    sBig[e] = ws[WS_EFFC + (size_t)b * N_MODULES * CONTENT_DIM + e];
  }
  __syncthreads();

  // 1/(||eff_c||^2 + EPS), one module per thread
  if (tid < N_MODULES) {
    float s = 0.f;
    for (int c = 0; c < CONTENT_DIM; ++c) {
      float v = sBig[tid * CONTENT_DIM + c];
      s += v * v;
    }
    ws[WS_INV + b * N_MODULES + tid] = 1.0f / (s + 1e-42f);
  }

  // h = relu([eff_p, eff_c] @ W1[m] + b1[m])
  for (int j = 0; j < 8; ++j) {
    int idx = tid + j * 256;
    int mm = idx >> 6;
    int o  = idx & 63;
    float acc = b1[mm * POINTER_DIM + o];
    const float* w1 = W1 + (size_t)mm * RIN_DIM * POINTER_DIM + o;
    for (int i = 0; i < POINTER_DIM; ++i)
      acc += sP[mm * POINTER_DIM + i] * w1[(size_t)i * POINTER_DIM];
    for (int i = 0; i < CONTENT_DIM; ++i)
      acc += sBig[mm * CONTENT_DIM + i] * w1[(size_t)(POINTER_DIM + i) * POINTER_DIM];
    sH[idx] = acc > 0.f ? acc : 0.f;
  }
  __syncthreads();

  // out_p = relu(h @ W2[m] + b2[m]) ; also store to d_out
  for (int j = 0; j < 8; ++j) {
    int idx = tid + j * 256;
    int mm = idx >> 6;
    int o  = idx & 63;
    float acc = b2[mm * POINTER_DIM + o];
    const float* w2 = W2 + (size_t)mm * POINTER_DIM * POINTER_DIM + o;
    for (int i = 0; i < POINTER_DIM; ++i)
      acc += sH[mm * POINTER_DIM + i] * w2[(size_t)i * POINTER_DIM];
    float v = acc > 0.f ? acc : 0.f;
    sOP[idx] = v;
    out[OUT_P_OFF + (size_t)b * N_MODULES * POINTER_DIM + idx] = v;
  }
  __syncthreads();

  // w_out[m][n] = relu(tanh(out_p[m] . out_p[n]))
  for (int j = 0; j < 4; ++j) {
    int idx = tid + j * 256;
    int mm = idx >> 5;
    int nn = idx & 31;
    float acc = 0.f;
    #pragma unroll 8
    for (int i = 0; i < POINTER_DIM; ++i)
      acc += sOP[mm * POINTER_DIM + i] * sOP[nn * POINTER_DIM + i];
    float w = tanhf(acc);
    sW[idx] = w > 0.f ? w : 0.f;
  }
  __syncthreads();

  // reuse sBig for out_c[b] (written by K2 into d_out)
  for (int j = 0; j < 32; ++j) {
    int e = tid + j * 256;
    sBig[e] = out[OUT_C_OFF + (size_t)b * N_MODULES * CONTENT_DIM + e];
  }
  __syncthreads();

  // eff_out_c[m][c] = sum_n w_out[m][n] * out_c[b,n,c]
  for (int j = 0; j < 32; ++j) {
    int idx = tid + j * 256;
    int mm = idx >> 8;
    int c  = idx & 255;
    float acc = 0.f;
    #pragma unroll
    for (int n = 0; n < N_MODULES; ++n)
      acc += sW[mm * N_MODULES + n] * sBig[n * CONTENT_DIM + c];
    ws[WS_EOUTC + (size_t)b * N_MODULES * CONTENT_DIM + idx] = acc;
  }
}

// ---------------------------------------------------------------------------
// K4: per (b,m) -> mems[b,m] = 0.5*memories[m] + (0.5*eff_out_c[c]*inv)*eff_c[d]
// Pure streaming: float4, prefetch the memories stream (L2-resident).
// ---------------------------------------------------------------------------
__global__ __launch_bounds__(256) void k4_mems(
    const float* __restrict__ memories,
    const float* __restrict__ ws,
    float* __restrict__ out) {
  const int bm  = blockIdx.x;       // b*32 + m
  const int b   = bm >> 5;
  const int m   = bm & 31;
  const int tid = threadIdx.x;
  __shared__ float sEo[CONTENT_DIM];  // 0.5 * inv * eff_out_c
  __shared__ float sEc[CONTENT_DIM];  // eff_c

  const float inv = ws[WS_INV + b * N_MODULES + m];
  sEo[tid] = ws[WS_EOUTC + ((size_t)b * N_MODULES + m) * CONTENT_DIM + tid] * 0.5f * inv;
  sEc[tid] = ws[WS_EFFC + ((size_t)b * N_MODULES + m) * CONTENT_DIM + tid];
  __syncthreads();

  const v4f* mem4 = (const v4f*)(memories + (size_t)m * CONTENT_DIM * CONTENT_DIM);
  v4f* out4 = (v4f*)(out + MEMS_OFF + (size_t)bm * CONTENT_DIM * CONTENT_DIM);
  const int total4 = CONTENT_DIM * CONTENT_DIM / 4;  // 16384

  for (int idx = tid; idx < total4; idx += 256) {
    if (idx + 2048 < total4) __builtin_prefetch(&mem4[idx + 2048], 0, 1);
    int row = idx >> 6;        // c
    int c4  = idx & 63;        // d/4
    v4f mv = mem4[idx];
    v4f ev = *(const v4f*)(&sEc[c4 * 4]);
    float s = sEo[row];
    v4f o;
    o.x = 0.5f * mv.x + s * ev.x;
    o.y = 0.5f * mv.y + s * ev.y;
    o.z = 0.5f * mv.z + s * ev.z;
    o.w = 0.5f * mv.w + s * ev.w;
    out4[idx] = o;
  }
}

// ---------------------------------------------------------------------------
extern "C" void kernel_launch(void* const* d_in, const int* in_sizes, int n_in,
                              void* d_out, int out_size, void* d_ws, size_t ws_size,
                              hipStream_t stream) {
  const float* pointers  = (const float*)d_in[0];  // [16,128,64]
  const float* contents  = (const float*)d_in[1];  // [16,128,256]
  const float* receivers = (const float*)d_in[2];  // [32,64]
  const float* memories  = (const float*)d_in[3];  // [32,256,256]
  const float* W1        = (const float*)d_in[4];  // [32,320,64]
  const float* b1        = (const float*)d_in[5];  // [32,64]
  const float* W2        = (const float*)d_in[6];  // [32,64,64]
  const float* b2        = (const float*)d_in[7];  // [32,64]
  float* out = (float*)d_out;
  float* ws  = (float*)d_ws;

  k1_gather <<<BATCH,               256, 0, stream>>>(pointers, contents, receivers, ws);
  k2_memread<<<N_MODULES,           512, 0, stream>>>(memories, ws, out);
  k3_route  <<<BATCH,               256, 0, stream>>>(W1, b1, W2, b2, ws, out);
  k4_mems   <<<BATCH * N_MODULES,   256, 0, stream>>>(memories, ws, out);
}